// GraphConvNet_61864708932310
// MI455X (gfx1250) — compile-verified
//
#include <hip/hip_runtime.h>
#include <hip/hip_bf16.h>
#include <math.h>

// ---------------------------------------------------------------------------
// GraphConvNet on MI455X (gfx1250, wave32).
//   Layer: h = relu( segsum(x[src]*w)[dst] @ Wrel + brel + x @ Wroot )
// GEMM-before-scatter (aggregation is linear) halves edge traffic; edge
// working set (y + agg, ~51 MB) is L2-resident (192 MB L2).
// Dense GEMMs: V_WMMA_F32_16X16X4_F32 (exact fp32), weights staged transposed
// in LDS (B fragments = ds_load_2addr_b64), A streamed as b64 global loads.
// ---------------------------------------------------------------------------

typedef __attribute__((ext_vector_type(2))) float v2f;
typedef __attribute__((ext_vector_type(8))) float v8f;

#define IN_DIM 128
#define HID_DIM 64
#define OUT_DIM 40

__device__ __forceinline__ int imin(int a, int b) { return a < b ? a : b; }

// ---------------------------------------------------------------------------
__global__ void zero_f32(float* __restrict__ p, int n) {
  int i = blockIdx.x * blockDim.x + threadIdx.x;
  int stride = gridDim.x * blockDim.x;
  for (; i < n; i += stride) p[i] = 0.0f;
}

// ---------------------------------------------------------------------------
// Dual GEMM with shared A-fragment:
//   Ya = X @ Wa            (n x NC)
//   Yb = X @ Wb + bias     (n x NC)
// X: n x K row-major.  Wa/Wb: K x NC row-major, staged TRANSPOSED in LDS as
// [col][k] (cols padded to CT*16 with zeros) so each lane's B pair is one
// contiguous 8-byte LDS read.  One wave per 16-row tile; k-loop outer,
// column tiles fully unrolled inner so one A load feeds 2*CT WMMAs.
//
// WMMA f32 16x16x4 operand layout (per ISA):
//   A (2 VGPR): lanes 0-15 -> M=lane,   K=kb+0 / kb+1
//               lanes 16-31 -> M=lane-16, K=kb+2 / kb+3
//   B mirrors with N=lane&15.
//   C/D (8 VGPR): VGPR r -> M=r (lanes 0-15) / M=r+8 (lanes 16-31), N=lane&15.
template <int K, int NC, int CT>
__global__ void __launch_bounds__(128)
gemm_dual_wmma(const float* __restrict__ X,
               const float* __restrict__ Wa,
               const float* __restrict__ Wb,
               const float* __restrict__ bias,
               float* __restrict__ Ya,
               float* __restrict__ Yb,
               int n) {
  constexpr int NCP = CT * 16;              // padded column count
  __shared__ float sWa[NCP * K];            // [col][k], transposed, zero-padded
  __shared__ float sWb[NCP * K];

  // Cooperative transpose-load of both weight matrices (once per block).
  for (int i = threadIdx.x; i < NCP * K; i += blockDim.x) {
    const int c = i / K;                    // K is a power of two -> shifts
    const int k = i - c * K;
    const bool ok = (c < NC);
    sWa[i] = ok ? Wa[k * NC + c] : 0.0f;
    sWb[i] = ok ? Wb[k * NC + c] : 0.0f;
  }
  __syncthreads();

  const int lane = threadIdx.x & 31;
  const int wave = threadIdx.x >> 5;
  const int tile = blockIdx.x * 4 + wave;
  const int m0 = tile * 16;
  if (m0 >= n) return;

  const int half = lane >> 4;               // 0: K=kb+0/1  1: K=kb+2/3
  const int l15 = lane & 15;
  const int mrow = imin(m0 + l15, n - 1);   // clamp (dup rows never stored)
  const float* __restrict__ xrow = X + (size_t)mrow * K;

  v8f accA[CT];
  v8f accB[CT];
#pragma unroll
  for (int ct = 0; ct < CT; ++ct) {
    accA[ct] = (v8f){0.f, 0.f, 0.f, 0.f, 0.f, 0.f, 0.f, 0.f};
    accB[ct] = (v8f){0.f, 0.f, 0.f, 0.f, 0.f, 0.f, 0.f, 0.f};
  }

  for (int kb = 0; kb < K; kb += 4) {
    const int ka = kb + 2 * half;
    const v2f a = *(const v2f*)(xrow + ka);               // global_load_b64
#pragma unroll
    for (int ct = 0; ct < CT; ++ct) {
      const int colp = ct * 16 + l15;
      const v2f bA = *(const v2f*)(sWa + colp * K + ka);  // ds_load (2addr) b64
      const v2f bB = *(const v2f*)(sWb + colp * K + ka);
      accA[ct] = __builtin_amdgcn_wmma_f32_16x16x4_f32(
          false, a, false, bA, (short)0, accA[ct], false, false);
      accB[ct] = __builtin_amdgcn_wmma_f32_16x16x4_f32(
          false, a, false, bB, (short)0, accB[ct], false, false);
    }
  }

  // Epilogue.  Fast path for full tiles (all tiles when 16 | n): straight-line
  // stores off a common base with immediate offsets; lane predicate (col<NC)
  // applied once per column tile, no per-row exec-mask churn.
  const int mbase = m0 + (half ? 8 : 0);
  if (m0 + 16 <= n) {
#pragma unroll
    for (int ct = 0; ct < CT; ++ct) {
      const int col = ct * 16 + l15;
      if (col < NC) {                      // uniform-per-lane predicate
        const float bv = bias[col];
        float* __restrict__ pa = Ya + (size_t)mbase * NC + col;
        float* __restrict__ pb = Yb + (size_t)mbase * NC + col;
#pragma unroll
        for (int r = 0; r < 8; ++r) {
          pa[(size_t)r * NC] = accA[ct][r];
          pb[(size_t)r * NC] = accB[ct][r] + bv;
        }
      }
    }
  } else {
#pragma unroll
    for (int ct = 0; ct < CT; ++ct) {
      const int col = ct * 16 + l15;
      if (col >= NC) continue;
      const float bv = bias[col];
#pragma unroll
      for (int r = 0; r < 8; ++r) {
        const int m = mbase + r;
        if (m < n) {
          Ya[(size_t)m * NC + col] = accA[ct][r];
          Yb[(size_t)m * NC + col] = accB[ct][r] + bv;
        }
      }
    }
  }
}

// ---------------------------------------------------------------------------
// Edge scatter, 64-wide rows: one wave per edge, each lane adds 2 floats.
// agg[dst, :] += y[src, :] * w[e].  y and agg are L2-resident (25.6 MB each).
__global__ void edge_scatter64(const long long* __restrict__ ei,
                               const float* __restrict__ ew,
                               const float* __restrict__ y,
                               float* __restrict__ agg, int E) {
  const long long gid = (long long)blockIdx.x * blockDim.x + threadIdx.x;
  const long long e = gid >> 5;
  if (e >= E) return;
  const int lane = (int)(gid & 31);
  const long long src = ei[e];
  const long long dst = ei[(long long)E + e];
  const float w = ew[e];
  const int d = lane * 2;
  const v2f yv = *(const v2f*)(y + src * 64 + d);       // global_load_b64
  float* ar = agg + dst * 64 + d;
  atomicAdd(&ar[0], yv.x * w);
  atomicAdd(&ar[1], yv.y * w);
}

// Edge scatter, 40-wide rows: 64 threads per edge, lanes 0..39 active.
__global__ void edge_scatter40(const long long* __restrict__ ei,
                               const float* __restrict__ ew,
                               const float* __restrict__ y,
                               float* __restrict__ agg, int E) {
  const long long gid = (long long)blockIdx.x * blockDim.x + threadIdx.x;
  const long long e = gid >> 6;
  const int d = (int)(gid & 63);
  if (e >= E || d >= 40) return;
  const long long src = ei[e];
  const long long dst = ei[(long long)E + e];
  const float w = ew[e];
  atomicAdd(&agg[dst * 40 + d], y[src * 40 + d] * w);
}

// ---------------------------------------------------------------------------
// h = relu(agg + root), in place into agg.
__global__ void relu_add(float* __restrict__ agg, const float* __restrict__ root,
                         int n) {
  int i = blockIdx.x * blockDim.x + threadIdx.x;
  int stride = gridDim.x * blockDim.x;
  for (; i < n; i += stride) {
    float v = agg[i] + root[i];
    agg[i] = v > 0.0f ? v : 0.0f;
  }
}

// ---------------------------------------------------------------------------
// log_softmax over 40 logits per node; one wave32 per node.
__global__ void log_softmax40(const float* __restrict__ agg,
                              const float* __restrict__ root,
                              float* __restrict__ out, int n) {
  const int lane = threadIdx.x & 31;
  const int wave = threadIdx.x >> 5;
  const int node = blockIdx.x * (blockDim.x >> 5) + wave;
  if (node >= n) return;

  const float* a = agg + (long long)node * 40;
  const float* r = root + (long long)node * 40;

  float z0 = a[lane] + r[lane];
  float z1 = (lane < 8) ? (a[32 + lane] + r[32 + lane]) : -INFINITY;

  float m = fmaxf(z0, z1);
#pragma unroll
  for (int off = 16; off > 0; off >>= 1)
    m = fmaxf(m, __shfl_xor(m, off, 32));

  float s = __expf(z0 - m) + ((lane < 8) ? __expf(z1 - m) : 0.0f);
#pragma unroll
  for (int off = 16; off > 0; off >>= 1)
    s += __shfl_xor(s, off, 32);

  const float lse = m + __logf(s);
  float* o = out + (long long)node * 40;
  o[lane] = z0 - lse;
  if (lane < 8) o[32 + lane] = z1 - lse;
}

// ---------------------------------------------------------------------------
extern "C" void kernel_launch(void* const* d_in, const int* in_sizes, int n_in,
                              void* d_out, int out_size, void* d_ws, size_t ws_size,
                              hipStream_t stream) {
  const float*     x      = (const float*)d_in[0];
  const long long* ei     = (const long long*)d_in[1];  // int64 [2, E]
  const float*     ew     = (const float*)d_in[2];
  const float*     Wrel1  = (const float*)d_in[3];
  const float*     brel1  = (const float*)d_in[4];
  const float*     Wroot1 = (const float*)d_in[5];
  const float*     Wrel2  = (const float*)d_in[6];
  const float*     brel2  = (const float*)d_in[7];
  const float*     Wroot2 = (const float*)d_in[8];

  const int N = in_sizes[0] / IN_DIM;
  const int E = in_sizes[2];

  // Workspace layout (fp32):
  float* y1   = (float*)d_ws;                 // N x 64 : x @ Wrel1
  float* r1   = y1   + (size_t)N * HID_DIM;   // N x 64 : x @ Wroot1 + brel1
  float* agg1 = r1   + (size_t)N * HID_DIM;   // N x 64 : scatter accum / h
  float* y2   = agg1 + (size_t)N * HID_DIM;   // N x 40 : h @ Wrel2
  float* r2   = y2   + (size_t)N * OUT_DIM;   // N x 40 : h @ Wroot2 + brel2
  float* agg2 = r2   + (size_t)N * OUT_DIM;   // N x 40 : scatter accum
  (void)ws_size;

  const int ZB = 256;
  zero_f32<<<(N * HID_DIM + ZB * 4 - 1) / (ZB * 4), ZB, 0, stream>>>(agg1, N * HID_DIM);
  zero_f32<<<(N * OUT_DIM + ZB * 4 - 1) / (ZB * 4), ZB, 0, stream>>>(agg2, N * OUT_DIM);

  const int tiles = (N + 15) / 16;
  const int blocks = (tiles + 3) / 4;          // 4 waves (128 threads) / block

  // ---- Layer 1 dense: y1 = x@Wrel1, r1 = x@Wroot1 + brel1 (WMMA) ----
  gemm_dual_wmma<IN_DIM, HID_DIM, 4><<<blocks, 128, 0, stream>>>(
      x, Wrel1, Wroot1, brel1, y1, r1, N);

  // ---- Layer 1 edge phase: agg1[dst] += y1[src] * w ----
  {
    const long long threads = (long long)E * 32;
    edge_scatter64<<<(int)((threads + 255) / 256), 256, 0, stream>>>(
        ei, ew, y1, agg1, E);
  }

  // ---- h = relu(agg1 + r1), in place ----
  relu_add<<<(N * HID_DIM + ZB * 4 - 1) / (ZB * 4), ZB, 0, stream>>>(
      agg1, r1, N * HID_DIM);

  // ---- Layer 2 dense: y2 = h@Wrel2, r2 = h@Wroot2 + brel2 (WMMA) ----
  gemm_dual_wmma<HID_DIM, OUT_DIM, 3><<<blocks, 128, 0, stream>>>(
      agg1, Wrel2, Wroot2, brel2, y2, r2, N);

  // ---- Layer 2 edge phase ----
  {
    const long long threads = (long long)E * 64;
    edge_scatter40<<<(int)((threads + 255) / 256), 256, 0, stream>>>(
        ei, ew, y2, agg2, E);
  }

  // ---- out = log_softmax(agg2 + r2) ----
  log_softmax40<<<(N + 7) / 8, 256, 0, stream>>>(agg2, r2, (float*)d_out, N);
}